// FWHT_34273839022653
// MI455X (gfx1250) — compile-verified
//
#include <hip/hip_runtime.h>
#include <hip/hip_bf16.h>

// FWHT N=16384 as Y = H128 * X * H128 (row viewed as 128x128 matrix),
// computed with fp32 WMMA (v_wmma_f32_16x16x4_f32) for exact fp32 precision.
// One workgroup (256 thr = 8 waves) per row; row lives entirely in LDS.
// HBM traffic is the theoretical minimum (one read + one write per element).

typedef float v2f __attribute__((ext_vector_type(2)));
typedef float v8f __attribute__((ext_vector_type(8)));

#define LDS_STRIDE 132   // 128+4 pad: conflict-free strided access, 16B-aligned rows

static __device__ __forceinline__ float xorf(float a, unsigned m) {
  return __uint_as_float(__float_as_uint(a) ^ m);
}

__global__ __launch_bounds__(256) void fwht16384_wmma_kernel(
    const float* __restrict__ x, float* __restrict__ out) {
  __shared__ float lds[128 * LDS_STRIDE];

  const int tid  = threadIdx.x;
  const int lane = tid & 31;
  const int wv   = tid >> 5;       // wave id 0..7 -> 16-row strip
  const int h    = lane >> 4;      // lane half
  const int n    = lane & 15;      // position within half
  const size_t base = (size_t)blockIdx.x * 16384;

  // ---- each wave loads ONLY its own 16-row strip (rows 16w..16w+15).
  // Pass-1 A-fragments read only these rows, so no entry barrier needed.
  {
    const float* src = x + base + wv * 2048;      // 16 rows * 128 = 8KB strip
    for (int i = 0; i < 16; ++i) {                // one full row per iteration
      float4 v = *(const float4*)(src + i * 128 + lane * 4);  // 512B/issue coalesced
      *(float4*)&lds[(wv * 16 + i) * LDS_STRIDE + lane * 4] = v;
    }
  }

  // Hadamard sign factorization over disjoint bit-fields of N = 16j + n:
  //   sign(K,N) = par((4k4)&n) ^ par((4k4)&(16j)) ^ par((2h+v)&n)
  const unsigned slo0    = (unsigned)(__popc((2 * h) & n) & 1) << 31; // K bits 1 (v=0)
  const unsigned oddmask = (unsigned)(n & 1) << 31;                   // K bit 0 flip (v=1)

  // pass0: U = X*H (store U^T)  ->  lds = H*X^T
  // pass1: V = (H X^T)*H (store V^T) -> lds = (H X^T H)^T = H X H = Y
  for (int pass = 0; pass < 2; ++pass) {
    v8f acc[8];
#pragma unroll
    for (int j = 0; j < 8; ++j) { v8f z = {}; acc[j] = z; }

    const int arow = (wv * 16 + n) * LDS_STRIDE;
    const int kk0  = 2 * h;
    // software-pipelined A fragment (16x4 f32: lane K = 4k4 + 2h {+0,+1} -> b64)
    v2f a_cur = *(const v2f*)&lds[arow + kk0];
    for (int k4 = 0; k4 < 32; ++k4) {
      v2f a_next = *(const v2f*)&lds[arow + 4 * ((k4 + 1) & 31) + kk0]; // prefetch (wraps, discarded)
      // base B pair for j = 0
      unsigned t0 = ((unsigned)(__popc((4 * k4) & n) & 1) << 31) ^ slo0;
      float b0x = __uint_as_float(0x3F800000u | t0);      // ±1.0f exact
      float b0y = xorf(b0x, oddmask);
#pragma unroll
      for (int j = 0; j < 8; ++j) {
        // scalar sign mask for N bits 4..6 (SALU): one v_xor per B register
        unsigned ujm = (unsigned)(__popc((4 * k4) & (16 * j)) & 1) << 31;
        v2f b;
        b.x = xorf(b0x, ujm);
        b.y = xorf(b0y, ujm);
        acc[j] = __builtin_amdgcn_wmma_f32_16x16x4_f32(
            false, a_cur, false, b, (short)0, acc[j], false, false);
      }
      a_cur = a_next;
    }
    __syncthreads();   // all waves done reading LDS before transposed overwrite

    // store D transposed: element (M = v+8h, N = n) of tile (i=wv, j)
    // consecutive v -> consecutive words, 16B aligned -> ds_store_b128 x2 per tile
#pragma unroll
    for (int j = 0; j < 8; ++j)
#pragma unroll
      for (int v = 0; v < 8; ++v)
        lds[(16 * j + n) * LDS_STRIDE + 16 * wv + v + 8 * h] = acc[j][v];
    __syncthreads();
  }

  // ---- lds holds Y; scaled coalesced float4 write-out ----
  const float scale = 1.0f / 128.0f;   // 1/sqrt(16384)
  for (int i = 0; i < 16; ++i) {
    int e = i * 1024 + tid * 4;
    int r = e >> 7, c = e & 127;
    float4 v = *(const float4*)&lds[r * LDS_STRIDE + c];
    v.x *= scale; v.y *= scale; v.z *= scale; v.w *= scale;
    *(float4*)(out + base + e) = v;
  }
}

extern "C" void kernel_launch(void* const* d_in, const int* in_sizes, int n_in,
                              void* d_out, int out_size, void* d_ws, size_t ws_size,
                              hipStream_t stream) {
  const float* x = (const float*)d_in[0];
  float* out = (float*)d_out;
  int rows = in_sizes[0] / 16384;   // 32*64 = 2048
  fwht16384_wmma_kernel<<<dim3(rows), dim3(256), 0, stream>>>(x, out);
  (void)n_in; (void)out_size; (void)d_ws; (void)ws_size;
}